// ComputeCDLoss_6476810682605
// MI455X (gfx1250) — compile-verified
//
#include <hip/hip_runtime.h>

typedef float v2f __attribute__((ext_vector_type(2)));
typedef float v8f __attribute__((ext_vector_type(8)));

__device__ __forceinline__ float half_wave_min16(float x) {
    // Butterfly min across each group of 16 lanes in a wave32.
    x = fminf(x, __shfl_xor(x, 8, 32));
    x = fminf(x, __shfl_xor(x, 4, 32));
    x = fminf(x, __shfl_xor(x, 2, 32));
    x = fminf(x, __shfl_xor(x, 1, 32));
    return x;
}

// Pack reference points into the exact V_WMMA_F32_16X16X4_F32 B-fragment
// layout: packed[b][tile][lane] (v2f). For point m in tile t = m/16, col
// c = m%16:  lane c      <- (gx, gy)        (K=0,1)
//            lane c + 16 <- (gz, |g|^2)     (K=2,3)
// So the min kernel's inner loop is a single b64 load per lane per tile.
__global__ __launch_bounds__(256) void cd_pack_kernel(
    const float* __restrict__ P,  // [B, M, 3]
    v2f* __restrict__ packed,     // [B, M/16, 32]
    int M, int B)
{
    const int idx = blockIdx.x * 256 + threadIdx.x;   // over B*M points
    if (idx >= B * M) return;
    const int b = idx / M;
    const int m = idx - b * M;

    const float* p = P + (size_t)idx * 3;
    const float x = p[0], y = p[1], z = p[2];
    const float g2 = x * x + y * y + z * z;

    const size_t base = ((size_t)b * (M >> 4) + (m >> 4)) * 32 + (m & 15);
    v2f lo; lo.x = x; lo.y = y;
    v2f hi; hi.x = z; hi.y = g2;
    packed[base]      = lo;
    packed[base + 16] = hi;
}

// For each of 16 query rows (one row-tile per wave), compute
//   minOut[b*N + row] = min_m ||Q[row] - R[m]||^2
// via implicit GEMM: A row = (-2x,-2y,-2z,1), B col = (gx,gy,gz,|g|^2)
// => WMMA yields -2 q.g + |g|^2 ; |q|^2 is added after the column-min.
__global__ __launch_bounds__(256) void cd_min_kernel(
    const float* __restrict__ Q,      // [B, N, 3] query points
    const v2f* __restrict__ packR,    // [B, M/16, 32] packed ref fragments
    float* __restrict__ minOut,       // [B*N]
    int N, int M, int B)
{
    const int lane = threadIdx.x & 31;
    const int wave = threadIdx.x >> 5;
    const int tilesPerBatch = N >> 4;
    const int tileIdx = blockIdx.x * 8 + wave;
    if (tileIdx >= B * tilesPerBatch) return;   // wave-uniform exit

    const int b  = tileIdx / tilesPerBatch;
    const int i0 = (tileIdx - b * tilesPerBatch) << 4;

    const float* Qb = Q + (size_t)b * N * 3;

    const int half   = lane >> 4;   // 0: holds K=0,1 ; 1: holds K=2,3
    const int rlocal = lane & 15;

    // A fragment (16x4 fp32): row = i0 + rlocal, row vector (-2x,-2y,-2z,1).
    const float* qp = Qb + (size_t)(i0 + rlocal) * 3;
    const float qx = qp[0], qy = qp[1], qz = qp[2];
    v2f afrag;
    afrag.x = half ? (-2.0f * qz) : (-2.0f * qx);
    afrag.y = half ?  1.0f        : (-2.0f * qy);

    v8f rowmin;
    #pragma unroll
    for (int v = 0; v < 8; ++v) rowmin[v] = 3.0e38f;

    const v2f* bp = packR + (size_t)b * (M >> 4) * 32 + lane;
    const int mTiles = M >> 4;

    #pragma unroll 4
    for (int t = 0; t < mTiles; ++t) {
        const v2f bfrag = bp[(size_t)t * 32];   // global_load_b64
        v8f c = {0.f, 0.f, 0.f, 0.f, 0.f, 0.f, 0.f, 0.f};
        c = __builtin_amdgcn_wmma_f32_16x16x4_f32(
                /*neg_a=*/false, afrag, /*neg_b=*/false, bfrag,
                /*c_mod=*/(short)0, c, /*reuse_a=*/false, /*reuse_b=*/false);
        #pragma unroll
        for (int v = 0; v < 8; ++v) rowmin[v] = fminf(rowmin[v], c[v]);
    }

    // C/D layout: lanes 0-15 VGPR v = row v; lanes 16-31 VGPR v = row v+8.
    // Reduce min over the 16 columns held across lanes of each half-wave.
    float rm[8];
    #pragma unroll
    for (int v = 0; v < 8; ++v) rm[v] = half_wave_min16(rowmin[v]);

    // Lane 0 finalizes rows i0..i0+7; lane 16 finalizes rows i0+8..i0+15,
    // adding the per-row |q|^2 that was factored out of the min.
    if (rlocal == 0) {
        const int base = half ? (i0 + 8) : i0;
        #pragma unroll
        for (int v = 0; v < 8; ++v) {
            const float* p = Qb + (size_t)(base + v) * 3;
            const float x = p[0], y = p[1], z = p[2];
            minOut[(size_t)b * N + base + v] = rm[v] + (x * x + y * y + z * z);
        }
    }
}

// Single-block reduction: loss = 0.5 * (mean(minA) + mean(minB))
__global__ __launch_bounds__(256) void cd_reduce_kernel(
    const float* __restrict__ minA, int na,
    const float* __restrict__ minB, int nb,
    float* __restrict__ out)
{
    __shared__ float sA[256];
    __shared__ float sB[256];
    float a = 0.f, bsum = 0.f;
    for (int i = threadIdx.x; i < na; i += 256) a    += minA[i];
    for (int i = threadIdx.x; i < nb; i += 256) bsum += minB[i];
    sA[threadIdx.x] = a;
    sB[threadIdx.x] = bsum;
    __syncthreads();
    for (int stride = 128; stride > 0; stride >>= 1) {
        if (threadIdx.x < stride) {
            sA[threadIdx.x] += sA[threadIdx.x + stride];
            sB[threadIdx.x] += sB[threadIdx.x + stride];
        }
        __syncthreads();
    }
    if (threadIdx.x == 0) {
        out[0] = 0.5f * (sA[0] / (float)na + sB[0] / (float)nb);
    }
}

extern "C" void kernel_launch(void* const* d_in, const int* in_sizes, int n_in,
                              void* d_out, int out_size, void* d_ws, size_t ws_size,
                              hipStream_t stream) {
    const float* recon = (const float*)d_in[0];   // [B, N, 3]
    const float* gt    = (const float*)d_in[1];   // [B, M, 3]

    const int B = 4;
    const int N = in_sizes[0] / (B * 3);
    const int M = in_sizes[1] / (B * 3);

    // Workspace layout (floats):
    //   min1     : B*N
    //   min2     : B*M
    //   packGT   : B*M*4   (v2f count B*M*2)
    //   packRec  : B*N*4
    float* wsf   = (float*)d_ws;
    float* min1  = wsf;
    float* min2  = min1 + (size_t)B * N;
    v2f*  packGT  = (v2f*)(min2 + (size_t)B * M);
    v2f*  packRec = packGT + (size_t)B * M * 2;

    // Pack both point sets into WMMA B-fragment layout.
    {
        const int nThreads = B * M;
        cd_pack_kernel<<<(nThreads + 255) / 256, 256, 0, stream>>>(gt, packGT, M, B);
    }
    {
        const int nThreads = B * N;
        cd_pack_kernel<<<(nThreads + 255) / 256, 256, 0, stream>>>(recon, packRec, N, B);
    }

    // Pass 1: nearest gt for each recon point.
    {
        const int tiles  = B * (N / 16);
        const int blocks = (tiles + 7) / 8;
        cd_min_kernel<<<blocks, 256, 0, stream>>>(recon, packGT, min1, N, M, B);
    }
    // Pass 2: nearest recon for each gt point (roles swapped, no atomics).
    {
        const int tiles  = B * (M / 16);
        const int blocks = (tiles + 7) / 8;
        cd_min_kernel<<<blocks, 256, 0, stream>>>(gt, packRec, min2, M, N, B);
    }

    cd_reduce_kernel<<<1, 256, 0, stream>>>(min1, B * N, min2, B * M,
                                            (float*)d_out);
}